// diff_extractor_59115929862503
// MI455X (gfx1250) — compile-verified
//
#include <hip/hip_runtime.h>
#include <stdint.h>

// Problem dims fixed by the reference:
// B=16, S=2048, T=128, hD=512, N=64
namespace {
constexpr int kB   = 16;
constexpr int kS   = 2048;
constexpr int kT   = 128;
constexpr int kHD  = 512;
constexpr int k2HD = 1024;
constexpr int kN   = 64;

// Async DMA-style 16B/lane copy: global -> LDS, tracked by ASYNCcnt.
// GV addressing mode: vdst = LDS byte address, vaddr = 64-bit global address.
__device__ __forceinline__ void async_copy16(uint32_t lds_byte_addr, const float* gptr) {
  uint64_t ga = (uint64_t)(uintptr_t)gptr;
  asm volatile("global_load_async_to_lds_b128 %0, %1, off"
               :: "v"(lds_byte_addr), "v"(ga)
               : "memory");
}

__device__ __forceinline__ void wait_async0() {
  asm volatile("s_wait_asynccnt 0" ::: "memory");
}

// One block (128 threads, 4 waves) per span. grid = (N, B, 3).
// z selects the span table in OUTPUT order: 0=para, 1=span(shell), 2=adu(x).
__global__ void __launch_bounds__(128)
span_diff_kernel(const float* __restrict__ word_reps,
                 const int* __restrict__ para_spans,
                 const int* __restrict__ shell_spans,
                 const int* __restrict__ x_spans,
                 float* __restrict__ out_base) {
  __shared__ float lds[4 * kHD];  // 8 KB staging: rows SF, EF, SB, EB

  const int n = blockIdx.x;
  const int b = blockIdx.y;
  const int z = blockIdx.z;

  const int* spans = (z == 0) ? para_spans : (z == 1) ? shell_spans : x_spans;
  const int* sp = spans + ((b * kN + n) * 3);   // block-uniform -> s_load
  const int e  = sp[0];
  const int s  = sp[1];
  const int en = sp[2];

  const uint64_t rowBase = (uint64_t)e * (uint64_t)kS;
  const float* rowSF = word_reps + (rowBase + (uint64_t)(s - 1)) * k2HD;         // fwd[e, s-1]
  const float* rowEF = word_reps + (rowBase + (uint64_t)en)      * k2HD;         // fwd[e, en]
  const float* rowSB = word_reps + (rowBase + (uint64_t)(en + 1)) * k2HD + kHD;  // bwd[e, en+1]
  const float* rowEB = word_reps + (rowBase + (uint64_t)s)       * k2HD + kHD;   // bwd[e, s]

  const int i = threadIdx.x * 4;  // float4 slice within hD (128 thr * 4 = 512)
  const uint32_t lbase = (uint32_t)(uintptr_t)(&lds[0]);

  async_copy16(lbase + (uint32_t)((0 * kHD + i) * 4), rowSF + i);
  async_copy16(lbase + (uint32_t)((1 * kHD + i) * 4), rowEF + i);
  async_copy16(lbase + (uint32_t)((2 * kHD + i) * 4), rowSB + i);
  async_copy16(lbase + (uint32_t)((3 * kHD + i) * 4), rowEB + i);

  wait_async0();  // each thread reads back only its own 16B slices: no barrier needed

  const float4 sf = *reinterpret_cast<const float4*>(&lds[0 * kHD + i]);
  const float4 ef = *reinterpret_cast<const float4*>(&lds[1 * kHD + i]);
  const float4 sb = *reinterpret_cast<const float4*>(&lds[2 * kHD + i]);
  const float4 eb = *reinterpret_cast<const float4*>(&lds[3 * kHD + i]);

  float* out = out_base +
      ((uint64_t)z * kB * kN + (uint64_t)b * kN + (uint64_t)n) * (uint64_t)(4 * kHD);

  const float4 d0 = make_float4(ef.x - sf.x, ef.y - sf.y, ef.z - sf.z, ef.w - sf.w);
  const float4 d1 = make_float4(eb.x - sb.x, eb.y - sb.y, eb.z - sb.z, eb.w - sb.w);

  *reinterpret_cast<float4*>(out + 0 * kHD + i) = d0;  // ef - sf
  *reinterpret_cast<float4*>(out + 1 * kHD + i) = d1;  // eb - sb
  *reinterpret_cast<float4*>(out + 2 * kHD + i) = sf;  // sf
  *reinterpret_cast<float4*>(out + 3 * kHD + i) = sb;  // sb
}

// topic[b, j] = topic_reps[b, lens[b]-1, j] for j < hD, else topic_reps[b, 0, j].
// One block per batch row; 256 threads * float4 = 1024 floats.
__global__ void __launch_bounds__(256)
topic_kernel(const float* __restrict__ topic_reps,
             const int* __restrict__ topic_lens,
             float* __restrict__ out) {
  const int b = blockIdx.x;
  const int j = threadIdx.x * 4;
  const int lastRow = topic_lens[b] - 1;   // block-uniform
  const int row = (j < kHD) ? lastRow : 0;
  const float4 v = *reinterpret_cast<const float4*>(
      topic_reps + ((uint64_t)b * kT + (uint64_t)row) * k2HD + j);
  *reinterpret_cast<float4*>(out + (uint64_t)b * k2HD + j) = v;
}
}  // namespace

extern "C" void kernel_launch(void* const* d_in, const int* in_sizes, int n_in,
                              void* d_out, int out_size, void* d_ws, size_t ws_size,
                              hipStream_t stream) {
  (void)in_sizes; (void)n_in; (void)out_size; (void)d_ws; (void)ws_size;

  const float* topic_reps  = (const float*)d_in[0];
  const float* word_reps   = (const float*)d_in[1];
  const int*   topic_lens  = (const int*)d_in[2];
  const int*   para_spans  = (const int*)d_in[3];
  const int*   x_spans     = (const int*)d_in[4];
  const int*   shell_spans = (const int*)d_in[5];

  float* out = (float*)d_out;
  const int topic_elems = kB * k2HD;  // 16384

  // Output layout: [topic | para | span(shell) | adu(x)]
  topic_kernel<<<kB, 256, 0, stream>>>(topic_reps, topic_lens, out);

  span_diff_kernel<<<dim3(kN, kB, 3), 128, 0, stream>>>(
      word_reps, para_spans, shell_spans, x_spans, out + topic_elems);
}